// Loss_PIP_5987184410889
// MI455X (gfx1250) — compile-verified
//
#include <hip/hip_runtime.h>
#include <math.h>

#define BB 16
#define CC 21
#define HH 256
#define WW 256
#define NBOX 20
#define HWSZ (HH*WW)
#define IGNORE_LBL 255

// blur kernels: fixed launch geometry -> compile-time-uniform tile loops
#define BLUR_BLOCKS 128
#define BLUR_THREADS 256
#define BLUR_WAVES (BLUR_BLOCKS * (BLUR_THREADS / 32))        // 1024
#define NTILES (BB * (HH/16) * (WW/16))                        // 4096
#define TILES_PER_WAVE (NTILES / BLUR_WAVES)                   // 4 (exact)

typedef __attribute__((ext_vector_type(2))) float v2f;
typedef __attribute__((ext_vector_type(8))) float v8f;

// -------- helpers --------

// branchless reflect (numpy 'reflect': -1 -> 1, n -> n-2)
__device__ __forceinline__ int reflect_idx(int i, int n) {
    i = (i < 0) ? -i : i;
    i = (i >= n) ? (2 * n - 2 - i) : i;
    return i;
}

// 1/sum of unnormalized 1-D gaussian (sigma=1, r=3), double to match numpy
__device__ __forceinline__ double gauss_inv_sum() {
    double s = 0.0;
    for (int i = 0; i < 7; i++) { double x = (double)(i - 3); s += exp(-0.5 * x * x); }
    return 1.0 / s;
}

// normalized gaussian weight g[d] for d in [0,6], else 0 (select, no branch)
__device__ __forceinline__ float gweight(int d, double inv_s) {
    double x = (double)(d - 3);
    float w = (float)(exp(-0.5 * x * x) * inv_s);
    return (d >= 0 && d <= 6) ? w : 0.0f;
}

__device__ __forceinline__ float block_reduce_sum(float v, float* red) {
    int tid = threadIdx.x;
    red[tid] = v;
    __syncthreads();
    for (int s = blockDim.x >> 1; s > 0; s >>= 1) {
        if (tid < s) red[tid] += red[tid + s];
        __syncthreads();
    }
    float r = red[0];
    __syncthreads();
    return r;
}

// -------- init scalars (must re-zero every call: graph replays) --------

__global__ void kInit(float* term_sum, int* hw_sum, unsigned int* ndmax,
                      float* ce_sum, float* gamma_sum) {
    int t = threadIdx.x;
    if (t == 0) { *term_sum = 0.0f; *hw_sum = 0; *ndmax = 0u; *ce_sum = 0.0f; }
    if (t < BB) gamma_sum[t] = 0.0f;
}

// -------- kernel A: per-box row/col softmax-prob max/min log terms --------

__global__ void kA_boxterms(const float* __restrict__ logits,
                            const int* __restrict__ bboxes,
                            float* term_sum, int* hw_sum) {
    __shared__ unsigned int cmax[64], cmin[64], rmax[64], rmin[64];
    __shared__ float red[256];
    int m = blockIdx.x;                 // box id, 0..BB*NBOX-1
    int b = m / NBOX;
    const int* bb = bboxes + m * 5;
    __builtin_prefetch(bb, 0, 0);       // global_prefetch_b8
    int x0 = bb[0], y0 = bb[1], x1 = bb[2], y1 = bb[3], cls = bb[4];
    if (cls == -1) return;              // invalid box: contributes nothing (uniform)
    int wbox = x1 - x0 + 1, hbox = y1 - y0 + 1;   // <= 64 each
    int tid = threadIdx.x;
    for (int i = tid; i < 64; i += blockDim.x) {
        cmax[i] = 0u;          rmax[i] = 0u;            // bits of +0.0
        cmin[i] = 0x7F7FFFFFu; rmin[i] = 0x7F7FFFFFu;   // bits of FLT_MAX
    }
    __syncthreads();
    const float* lg = logits + (size_t)b * CC * HWSZ;
    int npix = wbox * hbox;
    for (int i = tid; i < npix; i += blockDim.x) {
        int yy = y0 + i / wbox;
        int xx = x0 + i % wbox;
        int off = yy * WW + xx;
        float vv[CC];
        float mx = -3.4e38f;
        float vcls = 0.0f;
        #pragma unroll
        for (int c = 0; c < CC; c++) {
            float v = lg[(size_t)c * HWSZ + off];
            vv[c] = v;
            mx = fmaxf(mx, v);
            if (c == cls) vcls = v;
        }
        float se = 0.0f;
        #pragma unroll
        for (int c = 0; c < CC; c++) se += expf(vv[c] - mx);
        float p = expf(vcls - mx) / se;       // softmax prob, in (0,1)
        unsigned int pb = __float_as_uint(p); // positive floats: uint order == float order
        atomicMax(&cmax[xx - x0], pb);
        atomicMin(&cmin[xx - x0], pb);
        atomicMax(&rmax[yy - y0], pb);
        atomicMin(&rmin[yy - y0], pb);
    }
    __syncthreads();
    float acc = 0.0f;
    for (int i = tid; i < wbox; i += blockDim.x) {
        float cx = __uint_as_float(cmax[i]);
        float cn = __uint_as_float(cmin[i]);
        acc += 0.5f * logf(cx) + 0.5f * logf(1.0f - cn);   // ALPHA = 0.5
    }
    for (int i = tid; i < hbox; i += blockDim.x) {
        float rx = __uint_as_float(rmax[i]);
        float rn = __uint_as_float(rmin[i]);
        acc += 0.5f * logf(rx) + 0.5f * logf(1.0f - rn);
    }
    float tot = block_reduce_sum(acc, red);
    if (tid == 0) {
        atomicAdd(term_sum, tot);
        atomicAdd(hw_sum, wbox + hbox);
    }
}

// -------- kernel B: nop = 1 + #covering boxes; dis_pre = #perimeter hits --------

__global__ void kB_masks(const int* __restrict__ bboxes,
                         float* __restrict__ nop, float* __restrict__ dispre) {
    __shared__ int sb[NBOX * 5];
    int idx = blockIdx.x * blockDim.x + threadIdx.x;   // 0 .. BB*HWSZ-1
    int b = idx / HWSZ;                                 // uniform per block
    if (threadIdx.x < NBOX * 5) sb[threadIdx.x] = bboxes[b * NBOX * 5 + threadIdx.x];
    __syncthreads();
    int rem = idx % HWSZ;
    int y = rem / WW, x = rem % WW;
    int cfull = 0, cper = 0;
    #pragma unroll
    for (int n = 0; n < NBOX; n++) {
        int bx0 = sb[n*5+0], by0 = sb[n*5+1], bx1 = sb[n*5+2], by1 = sb[n*5+3], cl = sb[n*5+4];
        bool on = (cl != -1) && (x >= bx0) && (x <= bx1) && (y >= by0) && (y <= by1);
        bool edge = (x == bx0) || (x == bx1) || (y == by0) || (y == by1);
        cfull += on ? 1 : 0;
        cper  += (on && edge) ? 1 : 0;
    }
    nop[idx] = 1.0f + (float)cfull;
    dispre[idx] = (float)cper;
}

// -------- kernel C1: horizontal 7-tap gaussian via V_WMMA_F32_16X16X4_F32 --------
// out(16x16) = A(16x24) x Band(24x16), A[m][k] = src[y+m][x0-3+k] (reflect),
// Band[k][n] = g[k-n] for 0<=k-n<=6. Six K=4 accumulation steps.
// Fixed launch geometry -> compile-time trip counts -> EXEC all-ones at every WMMA.

__global__ void __launch_bounds__(BLUR_THREADS)
kC1_hblur(const float* __restrict__ src, float* __restrict__ tmp) {
    double inv_s = gauss_inv_sum();
    int lane = threadIdx.x & 31;
    int wave = threadIdx.x >> 5;
    int gw   = blockIdx.x * (BLUR_THREADS / 32) + wave;
    int mrow = lane & 15;     // A row / B col / D col
    int half = lane >> 4;
    const int TPI = (HH / 16) * (WW / 16);      // tiles per image
    #pragma unroll
    for (int it = 0; it < TILES_PER_WAVE; it++) {   // exactly 4, uniform
        int t   = gw + it * BLUR_WAVES;
        int b   = t / TPI;
        int rem = t % TPI;
        int yb  = (rem / (WW / 16)) * 16;
        int xb  = (rem % (WW / 16)) * 16;
        const float* sp = src + (size_t)b * HWSZ;
        v8f acc = {};
        #pragma unroll
        for (int c = 0; c < 6; c++) {
            int k0 = 4 * c + 2 * half;          // global K of this lane's pair
            // A: src row mrow, columns x0-3+k (reflect); zero the K=22,23 pad
            int cx0 = reflect_idx(xb - 3 + k0,     WW);
            int cx1 = reflect_idx(xb - 3 + k0 + 1, WW);
            v2f a;
            a.x = (k0     < 22) ? sp[(yb + mrow) * WW + cx0] : 0.0f;
            a.y = (k0 + 1 < 22) ? sp[(yb + mrow) * WW + cx1] : 0.0f;
            // B: banded gaussian, col n = mrow
            v2f bm;
            bm.x = gweight(k0     - mrow, inv_s);
            bm.y = gweight(k0 + 1 - mrow, inv_s);
            acc = __builtin_amdgcn_wmma_f32_16x16x4_f32(
                false, a, false, bm, (short)0, acc, false, false);
        }
        float* tp = tmp + (size_t)b * HWSZ;
        #pragma unroll
        for (int v = 0; v < 8; v++) {
            int mo = v + 8 * half;              // D: VGPR v holds rows v / v+8
            tp[(yb + mo) * WW + xb + mrow] = acc[v];
        }
    }
}

// -------- kernel C2: vertical pass (Band(16x24) x tmp(24x16)) + nd + global max ----

__global__ void __launch_bounds__(BLUR_THREADS)
kC2_vblur(const float* __restrict__ tmp, const float* __restrict__ nop,
          float* __restrict__ nd, unsigned int* ndmax_bits) {
    __shared__ float red[BLUR_THREADS];
    double inv_s = gauss_inv_sum();
    int lane = threadIdx.x & 31;
    int wave = threadIdx.x >> 5;
    int gw   = blockIdx.x * (BLUR_THREADS / 32) + wave;
    int mrow = lane & 15;
    int half = lane >> 4;
    const int TPI = (HH / 16) * (WW / 16);
    float lmax = 0.0f;
    #pragma unroll
    for (int it = 0; it < TILES_PER_WAVE; it++) {   // exactly 4, uniform
        int t   = gw + it * BLUR_WAVES;
        int b   = t / TPI;
        int rem = t % TPI;
        int yb  = (rem / (WW / 16)) * 16;
        int xb  = (rem % (WW / 16)) * 16;
        const float* tp = tmp + (size_t)b * HWSZ;
        const float* np = nop + (size_t)b * HWSZ;
        float*       op = nd  + (size_t)b * HWSZ;
        v8f acc = {};
        #pragma unroll
        for (int c = 0; c < 6; c++) {
            int k0 = 4 * c + 2 * half;
            // A: banded gaussian, row m = mrow (A[m][k] = g[k-m])
            v2f a;
            a.x = gweight(k0     - mrow, inv_s);
            a.y = gweight(k0 + 1 - mrow, inv_s);
            // B: tmp rows yb-3+k (reflect), col n = mrow; stray k=22,23 reads are
            // in-bounds after reflect and multiplied by zero band weights
            int ry0 = reflect_idx(yb - 3 + k0,     HH);
            int ry1 = reflect_idx(yb - 3 + k0 + 1, HH);
            v2f bm;
            bm.x = tp[ry0 * WW + xb + mrow];
            bm.y = tp[ry1 * WW + xb + mrow];
            acc = __builtin_amdgcn_wmma_f32_16x16x4_f32(
                false, a, false, bm, (short)0, acc, false, false);
        }
        #pragma unroll
        for (int v = 0; v < 8; v++) {
            int mo  = v + 8 * half;
            int off = (yb + mo) * WW + xb + mrow;
            float d  = acc[v] + 1.0f;           // dis = blur + 1
            float vn = np[off] * d;             // nd = nop * dis  (>= 1, positive)
            op[off] = vn;
            lmax = fmaxf(lmax, vn);
        }
    }
    // block max then single global atomic (positive floats: uint order works)
    int tid = threadIdx.x;
    red[tid] = lmax;
    __syncthreads();
    for (int s = blockDim.x >> 1; s > 0; s >>= 1) {
        if (tid < s) red[tid] = fmaxf(red[tid], red[tid + s]);
        __syncthreads();
    }
    if (tid == 0) atomicMax(ndmax_bits, __float_as_uint(red[0]));
}

// -------- kernel E: per-image Gamma sums --------

__global__ void kE_gammasum(const float* __restrict__ nd, const unsigned int* ndmax_bits,
                            float* gamma_sum) {
    __shared__ float red[256];
    int idx = blockIdx.x * blockDim.x + threadIdx.x;
    int b = idx / HWSZ;                              // uniform per block
    float ndm = __uint_as_float(*ndmax_bits);
    float x   = nd[idx] / ndm;
    float sig = 1.0f / (1.0f + expf(-x));
    float gam = (sig - 0.5f) * 1.0f + 1.0f;          // TAU = 1
    float tot = block_reduce_sum(gam, red);
    if (threadIdx.x == 0) atomicAdd(&gamma_sum[b], tot);
}

// -------- kernel F: CE * g_map weighted sum --------

__global__ void kF_ce(const float* __restrict__ logits, const int* __restrict__ labels,
                      const float* __restrict__ nd, const unsigned int* ndmax_bits,
                      const float* __restrict__ gamma_sum, float* ce_sum) {
    __shared__ float red[256];
    int idx = blockIdx.x * blockDim.x + threadIdx.x;
    int b   = idx / HWSZ;
    int off = idx % HWSZ;
    const float* lg = logits + (size_t)b * CC * HWSZ + off;
    int lbl = labels[idx];
    float vv[CC];
    float mx = -3.4e38f, vlbl = 0.0f;
    #pragma unroll
    for (int c = 0; c < CC; c++) {
        float v = lg[(size_t)c * HWSZ];
        vv[c] = v;
        mx = fmaxf(mx, v);
        if (c == lbl) vlbl = v;
    }
    float se = 0.0f;
    #pragma unroll
    for (int c = 0; c < CC; c++) se += expf(vv[c] - mx);
    float ce = (lbl == IGNORE_LBL) ? 0.0f : -(vlbl - mx - logf(se));
    float ndm = __uint_as_float(*ndmax_bits);
    float x   = nd[idx] / ndm;
    float sig = 1.0f / (1.0f + expf(-x));
    float gam = (sig - 0.5f) * 1.0f + 1.0f;
    float gmap = gam / gamma_sum[b] * (float)HWSZ;   // * (H*W)
    float tot = block_reduce_sum(gmap * ce, red);
    if (threadIdx.x == 0) atomicAdd(ce_sum, tot);
}

// -------- kernel G: final scalar --------

__global__ void kG_final(const float* term_sum, const int* hw_sum,
                         const float* ce_sum, float* out) {
    if (threadIdx.x == 0 && blockIdx.x == 0) {
        float num_rc  = 1e-5f + (float)(*hw_sum);
        float loss_rc = -(*term_sum);
        out[0] = loss_rc / num_rc + (*ce_sum) / (float)(BB * HWSZ);  // LAMB = 1
    }
}

// -------- launch --------

extern "C" void kernel_launch(void* const* d_in, const int* in_sizes, int n_in,
                              void* d_out, int out_size, void* d_ws, size_t ws_size,
                              hipStream_t stream) {
    const float* logits = (const float*)d_in[0];   // (16,21,256,256) f32
    const int*   bboxes = (const int*)d_in[1];     // (16,20,5) i32
    const int*   labels = (const int*)d_in[2];     // (16,256,256) i32
    float* out = (float*)d_out;

    char* ws = (char*)d_ws;
    float*        term_sum  = (float*)(ws + 0);
    int*          hw_sum    = (int*)(ws + 4);
    unsigned int* ndmax     = (unsigned int*)(ws + 8);
    float*        ce_sum    = (float*)(ws + 12);
    float*        gamma_sum = (float*)(ws + 16);       // 16 floats
    const size_t MAPB = (size_t)BB * HWSZ * sizeof(float);   // 4 MiB
    float* dispre = (float*)(ws + 256);
    float* nop    = (float*)(ws + 256 + MAPB);
    float* tmpb   = (float*)(ws + 256 + 2 * MAPB);
    float* nd     = (float*)(ws + 256 + 3 * MAPB);

    kInit<<<1, 32, 0, stream>>>(term_sum, hw_sum, ndmax, ce_sum, gamma_sum);
    kA_boxterms<<<BB * NBOX, 256, 0, stream>>>(logits, bboxes, term_sum, hw_sum);
    kB_masks<<<(BB * HWSZ) / 256, 256, 0, stream>>>(bboxes, nop, dispre);
    kC1_hblur<<<BLUR_BLOCKS, BLUR_THREADS, 0, stream>>>(dispre, tmpb);
    kC2_vblur<<<BLUR_BLOCKS, BLUR_THREADS, 0, stream>>>(tmpb, nop, nd, ndmax);
    kE_gammasum<<<(BB * HWSZ) / 256, 256, 0, stream>>>(nd, ndmax, gamma_sum);
    kF_ce<<<(BB * HWSZ) / 256, 256, 0, stream>>>(logits, labels, nd, ndmax, gamma_sum, ce_sum);
    kG_final<<<1, 32, 0, stream>>>(term_sum, hw_sum, ce_sum, out);
}